// FispBlock_50070728737523
// MI455X (gfx1250) — compile-verified
//
#include <hip/hip_runtime.h>
#include <stdint.h>

// EPG FISP scan, MI455X (gfx1250, wave32).
// One voxel per lane; full 3x21 complex state (63 x float2) register-resident
// across the P=100 sequential pulses. Dephase = register renaming/moves.
// Complex math written pairwise on float2 so the SLP vectorizer can emit
// v_pk_fma_f32 / v_dual_* (2x f32 per issue) for the dominant rotation loop.
// Per-pulse uniform RF matrix precomputed once into d_ws, then staged into
// LDS via the CDNA5 async path (GLOBAL_LOAD_ASYNC_TO_LDS + s_wait_asynccnt).

#define V_VOX 50000
#define NST   21
#define NP    100
#define REC   24          // floats per pulse record (16B-aligned records)

// record: [0]=m00  [1,2]=m01  [3,4]=m02  [5,6]=m10  [7]=m11 [8,9]=m12
//         [10,11]=m20 [12,13]=m21 [14]=m22 [15]=te [16]=tr-te  [17..23] pad
__global__ void epg_prep(const float* __restrict__ fa, const float* __restrict__ ph,
                         const float* __restrict__ te, const float* __restrict__ tr,
                         float* __restrict__ tbl) {
  int p = blockIdx.x * blockDim.x + threadIdx.x;
  if (p >= NP) return;
  float sina, cosa, sinp, cosp;
  __sincosf(fa[p], &sina, &cosa);
  __sincosf(ph[p], &sinp, &cosp);
  float cosa2 = 0.5f * (cosa + 1.0f);
  float sina2 = 1.0f - cosa2;
  float c2 = cosp * cosp - sinp * sinp;   // cos(2*phi)
  float s2 = 2.0f * sinp * cosp;          // sin(2*phi)
  float* r = tbl + p * REC;
  r[0]  = cosa2;                                      // m00
  r[1]  = sina2 * c2;        r[2]  =  sina2 * s2;     // m01 =  e^{+2i p} sina2
  r[3]  = sina * sinp;       r[4]  = -sina * cosp;    // m02 = -i e^{+i p} sina
  r[5]  = sina2 * c2;        r[6]  = -sina2 * s2;     // m10 =  e^{-2i p} sina2
  r[7]  = cosa2;                                      // m11
  r[8]  = sina * sinp;       r[9]  =  sina * cosp;    // m12 = +i e^{-i p} sina
  r[10] = -0.5f * sina * sinp; r[11] = -0.5f * sina * cosp; // m20
  r[12] = -0.5f * sina * sinp; r[13] =  0.5f * sina * cosp; // m21
  r[14] = cosa;                                       // m22
  r[15] = te[p];
  r[16] = tr[p] - te[p];
  r[17] = 0.f; r[18] = 0.f; r[19] = 0.f;
  r[20] = 0.f; r[21] = 0.f; r[22] = 0.f; r[23] = 0.f;
}

// acc += (cr + i*ci) * b, written pairwise so re/im pack into v_pk_fma_f32.
__device__ __forceinline__ float2 caxpy(float cr, float ci, float2 b, float2 acc) {
  acc.x = fmaf(cr, b.x, acc.x);
  acc.y = fmaf(cr, b.y, acc.y);
  acc.x = fmaf(-ci, b.y, acc.x);
  acc.y = fmaf(ci, b.x, acc.y);
  return acc;
}

__global__ void __launch_bounds__(256) epg_scan(
    const float* __restrict__ st_in, const float* __restrict__ t1,
    const float* __restrict__ t2,    const float* __restrict__ m0,
    const float* __restrict__ tbl,   float* __restrict__ out) {
  __shared__ float sp[NP * REC];
  const int tid = threadIdx.x;

  // CDNA5 async global->LDS copy of the 9.6KB per-pulse parameter table.
  for (int i = tid; i < NP * REC; i += 256) {
    unsigned lds = (unsigned)(uintptr_t)(&sp[i]);                  // low 32b = LDS byte addr
    unsigned long long ga = (unsigned long long)(uintptr_t)(tbl + i);
    asm volatile("global_load_async_to_lds_b32 %0, %1, off"
                 :: "v"(lds), "v"(ga) : "memory");
  }
  asm volatile("s_wait_asynccnt 0" ::: "memory");
  __syncthreads();

  const int v = blockIdx.x * 256 + tid;
  if (v < V_VOX) {
    float2 fp[NST], fm[NST], z[NST];
    const float2* si = (const float2*)st_in;
    const size_t b = (size_t)v * 3 * NST;
#pragma unroll
    for (int n = 0; n < NST; ++n) {
      fp[n] = si[b + n];
      fm[n] = si[b + NST + n];
      z[n]  = si[b + 2 * NST + n];
    }
    const float ni1 = -1.0f / t1[v];
    const float ni2 = -1.0f / t2[v];
    const float m0v = m0[v];
    float2* sig = (float2*)out + (size_t)V_VOX * 3 * NST;

#pragma unroll 1
    for (int p = 0; p < NP; ++p) {
      const float* r = sp + p * REC;
      const float m00 = r[0],  a1r = r[1],  a1i = r[2],  a2r = r[3],  a2i = r[4];
      const float b0r = r[5],  b0i = r[6],  m11 = r[7],  b2r = r[8],  b2i = r[9];
      const float c0r = r[10], c0i = r[11], c1r = r[12], c1i = r[13], m22 = r[14];
      const float te_s = r[15], trte = r[16];

      // RF rotation: uniform complex 3x3 applied to all N states (in place).
#pragma unroll
      for (int n = 0; n < NST; ++n) {
        float2 P = fp[n], M = fm[n], Z = z[n];
        float2 nP = make_float2(m00 * P.x, m00 * P.y);
        nP = caxpy(a1r, a1i, M, nP);
        nP = caxpy(a2r, a2i, Z, nP);
        float2 nM = make_float2(m11 * M.x, m11 * M.y);
        nM = caxpy(b0r, b0i, P, nM);
        nM = caxpy(b2r, b2i, Z, nM);
        float2 nZ = make_float2(m22 * Z.x, m22 * Z.y);
        nZ = caxpy(c0r, c0i, P, nZ);
        nZ = caxpy(c1r, c1i, M, nZ);
        fp[n] = nP; fm[n] = nM; z[n] = nZ;
      }
      // Relax over TE + T1 recovery on Z0.
      float e2 = __expf(te_s * ni2), e1 = __expf(te_s * ni1);
#pragma unroll
      for (int n = 0; n < NST; ++n) {
        fp[n].x *= e2; fp[n].y *= e2;
        fm[n].x *= e2; fm[n].y *= e2;
        z[n].x  *= e1; z[n].y  *= e1;
      }
      z[0].x += 1.0f - e1;
      // Acquisition.
      sig[(size_t)p * V_VOX + v] = make_float2(m0v * fp[0].x, m0v * fp[0].y);
      // Dephase (unit gradient): register shifts, F+[0] = conj(F-[1]).
      float2 h0 = make_float2(fm[1].x, -fm[1].y);
#pragma unroll
      for (int n = NST - 1; n >= 1; --n) fp[n] = fp[n - 1];
      fp[0] = h0;
#pragma unroll
      for (int n = 0; n < NST - 1; ++n) fm[n] = fm[n + 1];
      fm[NST - 1] = make_float2(0.f, 0.f);
      // Relax over TR-TE + recovery.
      e2 = __expf(trte * ni2); e1 = __expf(trte * ni1);
#pragma unroll
      for (int n = 0; n < NST; ++n) {
        fp[n].x *= e2; fp[n].y *= e2;
        fm[n].x *= e2; fm[n].y *= e2;
        z[n].x  *= e1; z[n].y  *= e1;
      }
      z[0].x += 1.0f - e1;
    }

    // Final state out (complex64 interleaved, [V,3,N]).
    float2* so = (float2*)out;
#pragma unroll
    for (int n = 0; n < NST; ++n) {
      so[b + n]           = fp[n];
      so[b + NST + n]     = fm[n];
      so[b + 2 * NST + n] = z[n];
    }
  }
}

extern "C" void kernel_launch(void* const* d_in, const int* in_sizes, int n_in,
                              void* d_out, int out_size, void* d_ws, size_t ws_size,
                              hipStream_t stream) {
  (void)in_sizes; (void)n_in; (void)out_size; (void)ws_size;
  const float* state = (const float*)d_in[0];  // [V,3,N] complex64
  const float* t1    = (const float*)d_in[1];
  const float* t2    = (const float*)d_in[2];
  const float* m0    = (const float*)d_in[3];
  const float* fa    = (const float*)d_in[4];
  const float* ph    = (const float*)d_in[5];
  const float* te    = (const float*)d_in[6];
  const float* tr    = (const float*)d_in[7];
  float* out = (float*)d_out;
  float* tbl = (float*)d_ws;                   // NP*REC*4 = 9600 bytes

  epg_prep<<<1, 128, 0, stream>>>(fa, ph, te, tr, tbl);
  epg_scan<<<(V_VOX + 255) / 256, 256, 0, stream>>>(state, t1, t2, m0, tbl, out);
}